// AdditiveAttention_66666482368820
// MI455X (gfx1250) — compile-verified
//
#include <hip/hip_runtime.h>

// Problem constants (from reference)
#define B_   4
#define LQ_  256
#define LK_  512
#define DQ_  512
#define DK_  512
#define DV_  512
#define H_   256
#define NEGV (-1e6f)

typedef __attribute__((ext_vector_type(2))) float    v2f;
typedef __attribute__((ext_vector_type(4))) float    v4f;
typedef __attribute__((ext_vector_type(8))) float    v8f;
typedef __attribute__((ext_vector_type(4))) unsigned u4v;
typedef __attribute__((ext_vector_type(8))) unsigned u8v;

// ---------------------------------------------------------------------------
// fp32 WMMA GEMM, 2x2 register tiling: each wave (32 threads) computes a
// 32x32 tile of C = A[MxK] @ Bm[KxN] (row-major) with V_WMMA_F32_16X16X4_F32.
// 4 WMMA per K-step, 6 VMEM per K-step (2x b64 A + 4x b32 B).
//
// ISA fragment layouts (CDNA5 ISA §7.12.2):
//   A 16x4 :  lane l -> M = l%16 ; VGPR0 = K=2*(l/16), VGPR1 = K=2*(l/16)+1
//   B 4x16 :  lane l -> N = l%16 ; VGPR0 = K=2*(l/16), VGPR1 = K=2*(l/16)+1
//   C 16x16:  lane l, VGPR v -> M = v + 8*(l/16), N = l%16
// ---------------------------------------------------------------------------
__global__ __launch_bounds__(32)
void wmma_gemm_f32(const float* __restrict__ A, const float* __restrict__ Bm,
                   float* __restrict__ C, int N, int K,
                   long aB, long bB, long cB) {
  const int lane = threadIdx.x;       // 0..31 (wave32)
  const int half = lane >> 4;         // 0 or 1
  const int l16  = lane & 15;
  const long bz  = blockIdx.z;
  A  += bz * aB;
  Bm += bz * bB;
  C  += bz * cB;

  const int m0 = blockIdx.x * 32;
  const int n0 = blockIdx.y * 32;

  const float* ap0 = A + (long)(m0 + l16) * K + half * 2;   // rows m0..m0+15
  const float* ap1 = ap0 + (long)16 * K;                    // rows m0+16..+31
  const float* bp0 = Bm + (long)(half * 2) * N + n0 + l16;  // cols n0..n0+15
  const float* bp1 = bp0 + 16;                              // cols n0+16..+31

  v8f acc00 = {}, acc01 = {}, acc10 = {}, acc11 = {};
#pragma unroll 4
  for (int k = 0; k < K; k += 4) {
    v2f a0 = *(const v2f*)ap0;        // 8B aligned (K even)
    v2f a1 = *(const v2f*)ap1;
    v2f b0, b1;
    b0.x = bp0[0]; b0.y = bp0[N];
    b1.x = bp1[0]; b1.y = bp1[N];
    acc00 = __builtin_amdgcn_wmma_f32_16x16x4_f32(false, a0, false, b0,
                                                  (short)0, acc00, false, false);
    acc01 = __builtin_amdgcn_wmma_f32_16x16x4_f32(false, a0, false, b1,
                                                  (short)0, acc01, false, false);
    acc10 = __builtin_amdgcn_wmma_f32_16x16x4_f32(false, a1, false, b0,
                                                  (short)0, acc10, false, false);
    acc11 = __builtin_amdgcn_wmma_f32_16x16x4_f32(false, a1, false, b1,
                                                  (short)0, acc11, false, false);
    ap0 += 4; ap1 += 4;
    bp0 += 4 * (long)N; bp1 += 4 * (long)N;
  }

  float* cp0 = C + (long)(m0 + half * 8) * N + n0 + l16;
  float* cp1 = cp0 + (long)16 * N;
#pragma unroll
  for (int v = 0; v < 8; ++v) {
    cp0[(long)v * N]      = acc00[v];
    cp0[(long)v * N + 16] = acc01[v];
    cp1[(long)v * N]      = acc10[v];
    cp1[(long)v * N + 16] = acc11[v];
  }
}

// ---------------------------------------------------------------------------
// CDNA5 V_TANH_F32 (trans pipe)
// ---------------------------------------------------------------------------
__device__ __forceinline__ float fast_tanh(float x) {
#if __has_builtin(__builtin_amdgcn_tanhf)
  return __builtin_amdgcn_tanhf(x);
#else
  float r;
  // v_nop satisfies the trans-op RAW hazard (1 indep op before use of result)
  asm("v_tanh_f32 %0, %1\n\tv_nop" : "=v"(r) : "v"(x));
  return r;
#endif
}

// ---------------------------------------------------------------------------
// Fused additive-attention scores + masked softmax.
// One 256-thread block per (q-row, batch).
//   scores[j] = sum_h wv[h]*tanh(q[h]+k[j,h]),  softmax over j (masked).
//
// kproj is staged h-chunk by h-chunk (512 rows x 32 h) into LDS by the
// Tensor Data Mover (tensor_load_to_lds, TENSORcnt), with TDM LDS padding
// (pad 4 dwords every 32 dwords -> 36-dword row stride) to spread rows
// across LDS banks. Inner loop is float4-vectorized (ds_load_b128).
// ---------------------------------------------------------------------------
#define KT_STRIDE 36   // dwords per LDS row: 32 data + 4 TDM pad

__global__ __launch_bounds__(256)
void attn_scores_softmax(const float* __restrict__ qproj,
                         const float* __restrict__ kproj,
                         const float* __restrict__ wv,
                         const int* __restrict__ valid_lens,
                         float* __restrict__ attn) {
  __shared__ __align__(128) float ktile[LK_ * KT_STRIDE];  // 72 KB
  __shared__ float qs[H_];
  __shared__ float wvs[H_];
  __shared__ float sc[LK_];
  __shared__ float red[256];

  const int t  = threadIdx.x;
  const int qi = blockIdx.x;
  const int b  = blockIdx.y;

  qs[t]  = qproj[((long)b * LQ_ + qi) * H_ + t];
  wvs[t] = wv[t];

  const unsigned lds_off = (unsigned)(unsigned long long)&ktile[0];
  const float* kbase = kproj + (long)b * LK_ * H_;

  float s0 = 0.f, s1 = 0.f;
  const float* kr0 = ktile + (long)t * KT_STRIDE;
  const float* kr1 = ktile + (long)(t + 256) * KT_STRIDE;

  for (int c = 0; c < H_ / 32; ++c) {
    const int hc = c * 32;

    __syncthreads();   // previous chunk fully consumed (also covers qs/wvs init)

    if (t < 32) {      // wave 0 issues the TDM DMA and waits TENSORcnt
      unsigned long long ga = (unsigned long long)(kbase + hc);
      u4v g0;
      g0[0] = 0x1u;                                   // count=1, user mode
      g0[1] = lds_off;                                // lds_addr (bytes)
      g0[2] = (unsigned)(ga & 0xffffffffu);           // global_addr[31:0]
      g0[3] = (unsigned)((ga >> 32) & 0x01ffffffu)    // global_addr[56:32]
              | 0x80000000u;                          // type=2 ("image")
      u8v g1;
      g1[0] = 0x07120000u;  // data_size=4B, pad_en, pad_interval=32dw, pad=4dw
      g1[1] = ((unsigned)H_ & 0xffffu) << 16;         // tensor_dim0 = 256 (lo16)
      g1[2] = ((unsigned)LK_ & 0xffffu) << 16;        // tensor_dim1 = 512 (lo16)
      g1[3] = 32u << 16;                              // tile_dim0 = 32 elems
      g1[4] = (unsigned)LK_;                          // tile_dim1 = 512 rows
      g1[5] = (unsigned)H_;                           // tensor_dim0_stride = 256
      g1[6] = 0u;
      g1[7] = 0u;
      asm volatile("tensor_load_to_lds %0, %1" :: "s"(g0), "s"(g1) : "memory");
      __builtin_amdgcn_s_wait_tensorcnt(0);
    }
    __syncthreads();   // LDS tile visible to all waves

#pragma unroll
    for (int hh = 0; hh < 32; hh += 4) {
      v4f q  = *(const v4f*)(qs  + hc + hh);
      v4f w  = *(const v4f*)(wvs + hc + hh);
      v4f k0 = *(const v4f*)(kr0 + hh);
      v4f k1 = *(const v4f*)(kr1 + hh);
      s0 = fmaf(w.x, fast_tanh(q.x + k0.x), s0);
      s1 = fmaf(w.x, fast_tanh(q.x + k1.x), s1);
      s0 = fmaf(w.y, fast_tanh(q.y + k0.y), s0);
      s1 = fmaf(w.y, fast_tanh(q.y + k1.y), s1);
      s0 = fmaf(w.z, fast_tanh(q.z + k0.z), s0);
      s1 = fmaf(w.z, fast_tanh(q.z + k1.z), s1);
      s0 = fmaf(w.w, fast_tanh(q.w + k0.w), s0);
      s1 = fmaf(w.w, fast_tanh(q.w + k1.w), s1);
    }
  }

  const int vl = valid_lens[b];
  s0 = (t       < vl) ? s0 : NEGV;
  s1 = (t + 256 < vl) ? s1 : NEGV;
  sc[t]       = s0;
  sc[t + 256] = s1;

  // block max
  red[t] = fmaxf(s0, s1);
  __syncthreads();
#pragma unroll
  for (int off = 128; off > 0; off >>= 1) {
    if (t < off) red[t] = fmaxf(red[t], red[t + off]);
    __syncthreads();
  }
  const float m = red[0];
  __syncthreads();

  // exp + block sum
  const float e0 = __expf(sc[t] - m);
  const float e1 = __expf(sc[t + 256] - m);
  red[t] = e0 + e1;
  __syncthreads();
#pragma unroll
  for (int off = 128; off > 0; off >>= 1) {
    if (t < off) red[t] += red[t + off];
    __syncthreads();
  }
  const float inv = 1.f / red[0];

  float* arow = attn + ((long)b * LQ_ + qi) * LK_;
  arow[t]       = e0 * inv;
  arow[t + 256] = e1 * inv;
}

// ---------------------------------------------------------------------------
extern "C" void kernel_launch(void* const* d_in, const int* in_sizes, int n_in,
                              void* d_out, int out_size, void* d_ws, size_t ws_size,
                              hipStream_t stream) {
  const float* queries    = (const float*)d_in[0];  // (B,LQ,DQ)
  const float* keys       = (const float*)d_in[1];  // (B,LK,DK)
  const float* values     = (const float*)d_in[2];  // (B,LK,DV)
  const int*   valid_lens = (const int*)  d_in[3];  // (B,)
  const float* Wq         = (const float*)d_in[4];  // (DQ,H)
  const float* Wk         = (const float*)d_in[5];  // (DK,H)
  const float* wv         = (const float*)d_in[6];  // (H,)
  float* out = (float*)d_out;                       // (B,LQ,DV)

  float* ws    = (float*)d_ws;
  float* qproj = ws;                                 // B*LQ*H  floats
  float* kproj = qproj + (long)B_ * LQ_ * H_;        // B*LK*H  floats
  float* attn  = kproj + (long)B_ * LK_ * H_;        // B*LQ*LK floats

  // q = Q @ Wq : weight shared across batch -> flatten (B*LQ, DQ)
  wmma_gemm_f32<<<dim3((B_ * LQ_) / 32, H_ / 32, 1), 32, 0, stream>>>(
      queries, Wq, qproj, H_, DQ_, 0, 0, 0);

  // k = K @ Wk : flatten (B*LK, DK)
  wmma_gemm_f32<<<dim3((B_ * LK_) / 32, H_ / 32, 1), 32, 0, stream>>>(
      keys, Wk, kproj, H_, DK_, 0, 0, 0);

  // scores + masked softmax (tanh-heavy stage; K staged via TDM into LDS)
  attn_scores_softmax<<<dim3(LQ_, B_), 256, 0, stream>>>(
      qproj, kproj, wv, valid_lens, attn);

  // out = attn @ V, batched over B
  wmma_gemm_f32<<<dim3(LQ_ / 32, DV_ / 32, B_), 32, 0, stream>>>(
      attn, values, out, DV_, LK_,
      (long)LQ_ * LK_, (long)LK_ * DV_, (long)LQ_ * DV_);
}